// AllophoneMapping_12953621364747
// MI455X (gfx1250) — compile-verified
//
#include <hip/hip_runtime.h>
#include <hip/hip_bf16.h>
#include <stdint.h>

// Problem constants (from reference): T=500, B=16, S=256, P=128, L=16
#define T_DIM 500
#define B_DIM 16
#define S_DIM 256
#define P_DIM 128
#define L_DIM 16
#define TT    10                 // timesteps per block tile (500 % 10 == 0)
#define SW    (S_DIM / 32)       // 8 mask words per (l,p) column

#define PAD_VALUE (-3.4028234663852886e+38f)  // np.finfo(float32).min == -FLT_MAX

// ---------------------------------------------------------------------------
// Kernel 1: bit-pack the allophone indicator matrices.
// masks[l][w][p] : bit i set  <=>  mats[l, w*32+i, p] != 0
// Grid: L*SW blocks of P threads; reads coalesced across p.
// ---------------------------------------------------------------------------
__global__ __launch_bounds__(P_DIM)
void build_masks_kernel(const float* __restrict__ mats, uint32_t* __restrict__ masks) {
    const int lw = blockIdx.x;          // l*SW + w
    const int l  = lw >> 3;
    const int w  = lw & 7;
    const int p  = threadIdx.x;

    const float* base = mats + (size_t)l * S_DIM * P_DIM + (size_t)(w * 32) * P_DIM + p;
    uint32_t bits = 0u;
    #pragma unroll
    for (int i = 0; i < 32; ++i) {
        bits |= (base[(size_t)i * P_DIM] != 0.0f) ? (1u << i) : 0u;
    }
    masks[(size_t)lw * P_DIM + p] = bits;
}

// ---------------------------------------------------------------------------
// Kernel 2: masked max-reduction.
//   out[t,b,p] = max over set bits s of logits[t,b,s], init PAD_VALUE.
// Grid: (T/TT, B), 128 threads (one per p). Logits tile staged to LDS via the
// CDNA5 async global->LDS copy path (ASYNCcnt), then bitmask-skip max.
// ---------------------------------------------------------------------------
template <bool USE_WS>
__global__ __launch_bounds__(P_DIM)
void allophone_max_kernel(const float* __restrict__ logits,
                          const int*   __restrict__ lang_ids,
                          const float* __restrict__ mats,
                          const uint32_t* __restrict__ masks,
                          float* __restrict__ out) {
    const int b  = blockIdx.y;
    const int t0 = blockIdx.x * TT;
    const int p  = threadIdx.x;
    const int lang = lang_ids[b];

    // --- per-thread bitmask for column (lang, p): 8 x 32 bits over s --------
    uint32_t m[SW];
    if (USE_WS) {
        #pragma unroll
        for (int w = 0; w < SW; ++w) {
            m[w] = masks[((size_t)lang * SW + w) * P_DIM + p];   // coalesced
        }
    } else {
        #pragma unroll
        for (int w = 0; w < SW; ++w) {
            const float* bp = mats + (size_t)lang * S_DIM * P_DIM
                                   + (size_t)(w * 32) * P_DIM + p;
            uint32_t bits = 0u;
            for (int i = 0; i < 32; ++i)
                bits |= (bp[(size_t)i * P_DIM] != 0.0f) ? (1u << i) : 0u;
            m[w] = bits;
        }
    }

    // --- stage logits tile [TT][S] into LDS with async global->LDS b128 ----
    __shared__ float slog[TT][S_DIM];

    // Row tt lives at gbase + tt*B*S floats; rows are 1 KB each.
    const float* gbase = logits + ((size_t)t0 * B_DIM + b) * S_DIM;

    // TT*S floats = 2560 = 640 float4 chunks; 128 threads -> 5 chunks each.
    #pragma unroll
    for (int iter = 0; iter < (TT * S_DIM / 4) / P_DIM; ++iter) {
        const int q  = iter * P_DIM + p;       // chunk id in [0, 640)
        const int tt = q >> 6;                 // 64 float4 chunks per row
        const int s4 = q & 63;
        const uint32_t goff   = (uint32_t)(((size_t)tt * B_DIM * S_DIM + (size_t)s4 * 4)
                                           * sizeof(float));
        const uint32_t ldsoff = (uint32_t)(uintptr_t)(&slog[tt][s4 * 4]);
        // GVS mode: LDS address taken from vdst; global address is SGPR base plus VGPR byte offset
        asm volatile("global_load_async_to_lds_b128 %0, %1, %2"
                     :
                     : "v"(ldsoff), "v"(goff), "s"(gbase)
                     : "memory");
    }
    // Drain ASYNCcnt before any wave signals the barrier.
    asm volatile("s_wait_asynccnt 0x0" ::: "memory");
    __syncthreads();

    // --- bitmask-skip max over set bits ------------------------------------
    const size_t obase = ((size_t)t0 * B_DIM + b) * P_DIM + p;
    #pragma unroll
    for (int tt = 0; tt < TT; ++tt) {
        float v = PAD_VALUE;
        #pragma unroll
        for (int w = 0; w < SW; ++w) {
            uint32_t bits = m[w];
            const float* row = &slog[tt][w * 32];
            while (bits) {
                const int i = __builtin_ctz(bits);
                bits &= bits - 1u;
                v = fmaxf(v, row[i]);
            }
        }
        out[obase + (size_t)tt * (B_DIM * P_DIM)] = v;   // coalesced over p
    }
}

// ---------------------------------------------------------------------------
// Host launch
// ---------------------------------------------------------------------------
extern "C" void kernel_launch(void* const* d_in, const int* in_sizes, int n_in,
                              void* d_out, int out_size, void* d_ws, size_t ws_size,
                              hipStream_t stream) {
    (void)in_sizes; (void)n_in; (void)out_size;

    const float* logits   = (const float*)d_in[0];   // [T,B,S] f32
    const int*   lang_ids = (const int*)d_in[1];     // [B] i32
    const float* mats     = (const float*)d_in[2];   // [L,S,P] f32 (0/1)
    // d_in[3] = allophone_mask (bool) is redundant: mask == (mats == 0)

    float* out = (float*)d_out;                      // [T,B,P] f32

    const size_t mask_bytes = (size_t)L_DIM * SW * P_DIM * sizeof(uint32_t); // 64 KB
    const bool use_ws = (ws_size >= mask_bytes) && (d_ws != nullptr);

    dim3 grid2(T_DIM / TT, B_DIM);   // 50 x 16 blocks
    dim3 block2(P_DIM);              // 128 threads = 4 wave32

    if (use_ws) {
        uint32_t* masks = (uint32_t*)d_ws;
        build_masks_kernel<<<dim3(L_DIM * SW), dim3(P_DIM), 0, stream>>>(mats, masks);
        allophone_max_kernel<true><<<grid2, block2, 0, stream>>>(
            logits, lang_ids, mats, masks, out);
    } else {
        allophone_max_kernel<false><<<grid2, block2, 0, stream>>>(
            logits, lang_ids, mats, nullptr, out);
    }
}